// Symmetrization_Net_28870770164234
// MI455X (gfx1250) — compile-verified
//
#include <hip/hip_runtime.h>

// CDNA5 / gfx1250. fp32 WMMA: D(16x16) = A(16x4) * B(4x16) + C.
typedef float v2f __attribute__((ext_vector_type(2)));
typedef float v8f __attribute__((ext_vector_type(8)));

// out[i,c] = 40320 * sum_j ( relu(x@W1+b1)@W2+b2 )[j,c]  for all i in 0..8
__global__ __launch_bounds__(64) void symnet_wmma_kernel(
    const float* __restrict__ x,    // [9, 512]
    const float* __restrict__ W1,   // [512, 32]
    const float* __restrict__ b1,   // [32]
    const float* __restrict__ W2,   // [32, 4]
    const float* __restrict__ b2,   // [4]
    float* __restrict__ out)        // [9, 4]
{
    __shared__ float h[16][32];   // relu'd hidden activations (rows 9..15 are padding)
    __shared__ float hs[32];      // column sums over the 9 valid rows
    __shared__ float s[4];        // final per-channel value (shared by all 9 out rows)

    const int tid   = threadIdx.x;
    const int wave  = tid >> 5;        // 0/1 -> which 16-column tile of W1
    const int lane  = tid & 31;
    const int nloc  = lane & 15;       // N within tile / M for A
    const int khalf = lane >> 4;       // 0: K={0,1}, 1: K={2,3} within each 16x4 step

    // A-matrix row (clamp padded rows 9..15 to row 8: stays in-bounds, result rows ignored)
    const int   m    = (nloc < 9) ? nloc : 8;
    const float* xrow = x + m * 512;
    const int   n    = wave * 16 + nloc;   // global hidden-unit index 0..31

    v8f c = {};
    #pragma unroll 4
    for (int k = 0; k < 512; k += 4) {
        const int ka = k + 2 * khalf;
        // A: 16x4 tile of x  (lane = row, VGPR pair = K)
        v2f a;
        a.x = xrow[ka];
        a.y = xrow[ka + 1];
        // B: 4x16 tile of W1 (lane = column, VGPR pair = K); W1 row-major [512,32]
        v2f b;
        b.x = W1[ka * 32 + n];
        b.y = W1[(ka + 1) * 32 + n];
        // 8 args: (neg_a, A, neg_b, B, c_mod, C, reuse_a, reuse_b)
        c = __builtin_amdgcn_wmma_f32_16x16x4_f32(
            false, a, false, b, (short)0, c, false, false);
    }

    // bias + ReLU, spill the 16x16 C tile: element r -> (row r + 8*khalf, col n)
    const float bias = b1[n];
    #pragma unroll
    for (int r = 0; r < 8; ++r) {
        float v = c[r] + bias;
        h[r + 8 * khalf][n] = (v > 0.0f) ? v : 0.0f;
    }
    __syncthreads();

    // hs[n] = sum over the 9 valid rows
    if (tid < 32) {
        float acc = 0.0f;
        #pragma unroll
        for (int j = 0; j < 9; ++j) acc += h[j][tid];
        hs[tid] = acc;
    }
    __syncthreads();

    // s[c] = 40320 * ( hs . W2[:,c] + 9*b2[c] )   (sum of 9 identical b2 adds)
    if (tid < 4) {
        float acc = 9.0f * b2[tid];
        #pragma unroll
        for (int nn = 0; nn < 32; ++nn) acc += hs[nn] * W2[nn * 4 + tid];
        s[tid] = 40320.0f * acc;
    }
    __syncthreads();

    // out is [9,4] with all rows identical
    if (tid < 36) out[tid] = s[tid & 3];
}

extern "C" void kernel_launch(void* const* d_in, const int* in_sizes, int n_in,
                              void* d_out, int out_size, void* d_ws, size_t ws_size,
                              hipStream_t stream) {
    (void)in_sizes; (void)n_in; (void)d_ws; (void)ws_size; (void)out_size;
    const float* x  = (const float*)d_in[0];
    const float* W1 = (const float*)d_in[1];
    const float* b1 = (const float*)d_in[2];
    const float* W2 = (const float*)d_in[3];
    const float* b2 = (const float*)d_in[4];
    // d_in[5] = perms table: provably unneeded (each row appears 8! times per slot)
    symnet_wmma_kernel<<<1, 64, 0, stream>>>(x, W1, b1, W2, b2, (float*)d_out);
}